// LiGhT_23021024706983
// MI455X (gfx1250) — compile-verified
//
#include <hip/hip_runtime.h>
#include <cstdint>

// ---------------- problem constants ----------------
#define N_NODES  50000
#define E_EDGES  600000
#define D_MODEL  128
#define N_HEADS  8
#define L_PATH   5
#define DT_DIM   32
#define DFF_DIM  512
#define HD_DIM   16
#define NLAYERS  2
#define EPS_LN   1e-5f

typedef float v2f __attribute__((ext_vector_type(2)));
typedef float v8f __attribute__((ext_vector_type(8)));

__device__ __forceinline__ float gelu_exact(float x) {
  return 0.5f * x * (1.0f + erff(x * 0.70710678118654752440f));
}

// gfx1250 async copy: global -> LDS, 16 bytes per lane, tracked by ASYNCcnt.
__device__ __forceinline__ void async_ld16(unsigned lds_addr, const float* gaddr) {
  asm volatile("global_load_async_to_lds_b128 %0, %1, off"
               :: "v"(lds_addr), "v"(gaddr) : "memory");
}
__device__ __forceinline__ unsigned lds_off(const void* p) {
  // generic pointer to LDS: low 32 bits are the wave-relative LDS byte address
  return (unsigned)(uintptr_t)p;
}

// ---------------------------------------------------------------------------
// Generic fp32 WMMA GEMM:  C[M x Nc] = act(A[M x K] @ W[K x Nc] + bias) + res
// Block tile 64(M) x 32(N), 8 waves, each wave one 16x16 tile.
// Interior blocks: double-buffered GLOBAL_LOAD_ASYNC_TO_LDS_B128 pipeline.
// M-edge block (at most one per GEMM): guarded synchronous staging.
// Requires K % 32 == 0 and Nc % 32 == 0 (true for all GEMMs here).
// ---------------------------------------------------------------------------
#define LA_STRIDE 36   // 16B-aligned rows; 36m mod 64 distinct -> conflict-free
#define LB_STRIDE 48   // 2-row offset = 32 banks -> half-wave sets disjoint

__device__ __forceinline__ v8f wmma_tile(const float (*As)[LA_STRIDE],
                                         const float (*Bs)[LB_STRIDE],
                                         int arow, int bcol, int half, v8f acc) {
#pragma unroll
  for (int kk = 0; kk < 32; kk += 4) {
    int ka = kk + 2 * half;
    v2f a, b;
    a[0] = As[arow][ka + 0];
    a[1] = As[arow][ka + 1];
    b[0] = Bs[ka + 0][bcol];
    b[1] = Bs[ka + 1][bcol];
    acc = __builtin_amdgcn_wmma_f32_16x16x4_f32(
        false, a, false, b, (short)0, acc, false, false);
  }
  return acc;
}

__global__ __launch_bounds__(256) void gemm_f32_wmma(
    const float* __restrict__ A, const float* __restrict__ W,
    const float* __restrict__ bias, const float* __restrict__ res,
    float* __restrict__ C, int M, int K, int Nc, int do_gelu)
{
  __shared__ float lA[2][64][LA_STRIDE];
  __shared__ float lB[2][32][LB_STRIDE];

  const int tid  = threadIdx.x;
  const int wave = tid >> 5;       // 0..7
  const int lane = tid & 31;
  const int wm   = wave >> 1;      // 0..3  : 16-row tile within block
  const int wn   = wave & 1;       // 0..1  : 16-col tile within block
  const int lo   = lane & 15;
  const int half = lane >> 4;

  const int bm = blockIdx.x * 64;
  const int bn = blockIdx.y * 32;
  const int nglob = bn + wn * 16 + lo;   // this lane's output column
  const int arow = wm * 16 + lo;
  const int bcol = wn * 16 + lo;

  float bv = bias ? bias[nglob] : 0.0f;
  v8f acc;
#pragma unroll
  for (int i = 0; i < 8; ++i) acc[i] = bv;

  // staging coordinates
  const int ar0 = (tid + 0)   >> 3, ac0 = (tid + 0)   & 7;   // A piece 0
  const int ar1 = (tid + 256) >> 3, ac1 = (tid + 256) & 7;   // A piece 1
  const int br  = tid >> 3,         bc  = tid & 7;           // B piece

  if (bm + 64 <= M) {
    // ---------------- async double-buffered pipeline ----------------
    const int nst = K >> 5;
    // prologue: stage 0 -> buffer 0
    async_ld16(lds_off(&lA[0][ar0][ac0 * 4]), A + (size_t)(bm + ar0) * K + ac0 * 4);
    async_ld16(lds_off(&lA[0][ar1][ac1 * 4]), A + (size_t)(bm + ar1) * K + ac1 * 4);
    async_ld16(lds_off(&lB[0][br][bc * 4]),   W + (size_t)br * Nc + bn + bc * 4);

    for (int s = 0; s < nst; ++s) {
      const int p = s & 1;
      if (s + 1 < nst) {
        const int k1 = (s + 1) << 5;
        async_ld16(lds_off(&lA[p ^ 1][ar0][ac0 * 4]),
                   A + (size_t)(bm + ar0) * K + k1 + ac0 * 4);
        async_ld16(lds_off(&lA[p ^ 1][ar1][ac1 * 4]),
                   A + (size_t)(bm + ar1) * K + k1 + ac1 * 4);
        async_ld16(lds_off(&lB[p ^ 1][br][bc * 4]),
                   W + (size_t)(k1 + br) * Nc + bn + bc * 4);
        asm volatile("s_wait_asynccnt 0x3" ::: "memory");  // stage s done
      } else {
        asm volatile("s_wait_asynccnt 0x0" ::: "memory");
      }
      __syncthreads();
      acc = wmma_tile(lA[p], lB[p], arow, bcol, half, acc);
      __syncthreads();  // buffer p free for stage s+2
    }
  } else {
    // ---------------- M-edge block: guarded synchronous staging ----------------
    for (int k0 = 0; k0 < K; k0 += 32) {
#pragma unroll
      for (int i = 0; i < 8; ++i) {
        int id = tid + i * 256;
        int r = id >> 5, c = id & 31;
        int gr = bm + r;
        lA[0][r][c] = (gr < M) ? A[(size_t)gr * K + k0 + c] : 0.0f;
      }
      {
        float4 wv = *reinterpret_cast<const float4*>(
            W + (size_t)(k0 + br) * Nc + bn + bc * 4);
        *reinterpret_cast<float4*>(&lB[0][br][bc * 4]) = wv;
      }
      __syncthreads();
      acc = wmma_tile(lA[0], lB[0], arow, bcol, half, acc);
      __syncthreads();
    }
  }

  // epilogue: VGPR i -> row (wm*16 + 8*half + i), col nglob
#pragma unroll
  for (int i = 0; i < 8; ++i) {
    int m = bm + wm * 16 + 8 * half + i;
    if (m < M) {
      float v = acc[i];
      if (do_gelu) v = gelu_exact(v);
      if (res) v += res[(size_t)m * Nc + nglob];
      C[(size_t)m * Nc + nglob] = v;
    }
  }
}

// ---------------- helper kernels ----------------

__global__ void fill_f32(float* p, float v, long n) {
  long i = (long)blockIdx.x * blockDim.x + threadIdx.x;
  if (i < n) p[i] = v;
}

__global__ void copy_f32(const float* __restrict__ a, float* __restrict__ b, long n) {
  long i = (long)blockIdx.x * blockDim.x + threadIdx.x;
  if (i < n) b[i] = a[i];
}

// dist_attn has only 8 distinct rows: compute the 8x8 table in one block.
__global__ void dist8_kernel(
    const float* __restrict__ path_len_emb,   // (L+1) x D
    const float* __restrict__ vpe,            // D
    const float* __restrict__ sle,            // D
    const float* __restrict__ W1, const float* __restrict__ b1,  // DxD, D
    const float* __restrict__ W2, const float* __restrict__ b2,  // DxH, H
    float* __restrict__ dist8)                // 8 x H
{
  __shared__ float rows[8][D_MODEL];
  __shared__ float hid[8][D_MODEL];
  int t = threadIdx.x;  // 128 threads
  for (int r = 0; r < 8; ++r) {
    float v;
    if (r < L_PATH + 1)      v = path_len_emb[r * D_MODEL + t];
    else if (r == 6)         v = vpe[t];
    else                     v = sle[t];
    rows[r][t] = v;
  }
  __syncthreads();
  for (int r = 0; r < 8; ++r) {
    float a = b1[t];
    for (int k = 0; k < D_MODEL; ++k) a += rows[r][k] * W1[k * D_MODEL + t];
    hid[r][t] = gelu_exact(a);
  }
  __syncthreads();
  if (t < 8 * N_HEADS) {
    int r = t >> 3, h = t & 7;
    float a = b2[h];
    for (int k = 0; k < D_MODEL; ++k) a += hid[r][k] * W2[k * N_HEADS + h];
    dist8[r * N_HEADS + h] = a;
  }
}

// per-edge: which of the 8 dist rows applies (priority: self-loop > virtual > count)
__global__ void sel_kernel(const int* __restrict__ path,
                           const int* __restrict__ vp, const int* __restrict__ sl,
                           int* __restrict__ sel)
{
  int e = blockIdx.x * blockDim.x + threadIdx.x;
  if (e >= E_EDGES) return;
  int cnt = 0;
#pragma unroll
  for (int l = 0; l < L_PATH; ++l) cnt += (path[e * L_PATH + l] >= 0) ? 1 : 0;
  int s = cnt;
  if (vp[e] == 1) s = 6;
  if (sl[e] == 1) s = 7;
  sel[e] = s;
}

// path_h[e][t] = sum_l t_all[l][path(e,l) or N][t] / count(e)
__global__ void path_gather_kernel(const int* __restrict__ path,
                                   const float* __restrict__ t_all,
                                   float* __restrict__ path_h)
{
  int e = blockIdx.x * 8 + (threadIdx.x >> 5);
  int t = threadIdx.x & 31;
  if (e >= E_EDGES) return;
  int cnt = 0;
  float s = 0.0f;
#pragma unroll
  for (int l = 0; l < L_PATH; ++l) {
    int pv = path[e * L_PATH + l];
    int idx = (pv < 0) ? N_NODES : pv;
    cnt += (pv >= 0) ? 1 : 0;
    s += t_all[(size_t)l * (N_NODES + 1) * DT_DIM + (size_t)idx * DT_DIM + t];
  }
  float c = (float)(cnt > 0 ? cnt : 1);
  path_h[(size_t)e * DT_DIM + t] = s / c;
}

// path_attn[e][h] = ph1[e] . W2[:,h] + b2[h]   (N=8 too narrow for a tile)
__global__ void path_attn_kernel(const float* __restrict__ ph1,
                                 const float* __restrict__ W2,
                                 const float* __restrict__ b2,
                                 float* __restrict__ path_attn)
{
  long i = (long)blockIdx.x * blockDim.x + threadIdx.x;
  if (i >= (long)E_EDGES * N_HEADS) return;
  long e = i >> 3;
  int h = (int)(i & 7);
  float a = b2[h];
  const float* row = ph1 + (size_t)e * DT_DIM;
#pragma unroll
  for (int k = 0; k < DT_DIM; ++k) a += row[k] * W2[k * N_HEADS + h];
  path_attn[i] = a;
}

// LayerNorm, one wave per row (D=128 -> float4 per lane), biased variance.
__global__ __launch_bounds__(256) void ln_kernel(
    const float* __restrict__ in, const float* __restrict__ g,
    const float* __restrict__ b, float* __restrict__ out, int M)
{
  int row = blockIdx.x * 8 + (threadIdx.x >> 5);
  int lane = threadIdx.x & 31;
  if (row >= M) return;
  const float4* ip = reinterpret_cast<const float4*>(in + (size_t)row * D_MODEL);
  float4 x = ip[lane];
  float s  = x.x + x.y + x.z + x.w;
  float sq = x.x * x.x + x.y * x.y + x.z * x.z + x.w * x.w;
#pragma unroll
  for (int m = 1; m < 32; m <<= 1) {
    s  += __shfl_xor(s,  m, 32);
    sq += __shfl_xor(sq, m, 32);
  }
  float mu  = s * (1.0f / D_MODEL);
  float var = sq * (1.0f / D_MODEL) - mu * mu;
  float r = rsqrtf(var + EPS_LN);
  int c = lane * 4;
  float4 o;
  o.x = (x.x - mu) * r * g[c + 0] + b[c + 0];
  o.y = (x.y - mu) * r * g[c + 1] + b[c + 1];
  o.z = (x.z - mu) * r * g[c + 2] + b[c + 2];
  o.w = (x.w - mu) * r * g[c + 3] + b[c + 3];
  reinterpret_cast<float4*>(out + (size_t)row * D_MODEL)[lane] = o;
}

// logits[e][h] = scale * q[src].k[dst] + dist8[sel[e]][h] + path_attn[e][h]
__global__ void logits_kernel(const float* __restrict__ qkv,
                              const int* __restrict__ src, const int* __restrict__ dst,
                              const int* __restrict__ sel,
                              const float* __restrict__ dist8,
                              const float* __restrict__ path_attn,
                              float* __restrict__ logits)
{
  long i = (long)blockIdx.x * blockDim.x + threadIdx.x;
  if (i >= (long)E_EDGES * N_HEADS) return;
  long e = i >> 3;
  int h = (int)(i & 7);
  const float* q = qkv + (size_t)src[e] * (3 * D_MODEL) + h * HD_DIM;
  const float* k = qkv + (size_t)dst[e] * (3 * D_MODEL) + D_MODEL + h * HD_DIM;
  float a = 0.0f;
#pragma unroll
  for (int d = 0; d < HD_DIM; ++d) a += q[d] * k[d];
  const float scale = 0.08838834764831845f;  // D^-0.5
  logits[i] = a * scale + dist8[sel[e] * N_HEADS + h] + path_attn[i];
}

__device__ __forceinline__ void atomicMaxF(float* addr, float val) {
  if (val >= 0.0f) atomicMax((int*)addr, __float_as_int(val));
  else             atomicMin((unsigned int*)addr, __float_as_uint(val));
}

__global__ void segmax_kernel(const float* __restrict__ logits,
                              const int* __restrict__ dst, float* __restrict__ m)
{
  long i = (long)blockIdx.x * blockDim.x + threadIdx.x;
  if (i >= (long)E_EDGES * N_HEADS) return;
  long e = i >> 3;
  int h = (int)(i & 7);
  atomicMaxF(&m[(size_t)dst[e] * N_HEADS + h], logits[i]);
}

// ex = exp(logit - m[dst]) written in place over logits; den += ex
__global__ void exps_kernel(float* __restrict__ logits,
                            const int* __restrict__ dst,
                            const float* __restrict__ m, float* __restrict__ den)
{
  long i = (long)blockIdx.x * blockDim.x + threadIdx.x;
  if (i >= (long)E_EDGES * N_HEADS) return;
  long e = i >> 3;
  int h = (int)(i & 7);
  float ex = __expf(logits[i] - m[(size_t)dst[e] * N_HEADS + h]);
  logits[i] = ex;
  atomicAdd(&den[(size_t)dst[e] * N_HEADS + h], ex);
}

// agg[dst][d] += v[src][d] * ex/den  (h = d/16)
__global__ void agg_kernel(const float* __restrict__ logits,
                           const float* __restrict__ den,
                           const float* __restrict__ qkv,
                           const int* __restrict__ src, const int* __restrict__ dst,
                           float* __restrict__ agg)
{
  long i = (long)blockIdx.x * blockDim.x + threadIdx.x;
  if (i >= (long)E_EDGES * D_MODEL) return;
  long e = i >> 7;
  int d = (int)(i & 127);
  int h = d >> 4;
  int de = dst[e];
  float sa = logits[e * N_HEADS + h] / den[(size_t)de * N_HEADS + h];
  float v  = qkv[(size_t)src[e] * (3 * D_MODEL) + 2 * D_MODEL + d];
  atomicAdd(&agg[(size_t)de * D_MODEL + d], v * sa);
}

// ---------------- host-side orchestration ----------------

static inline void gemm(const float* A, const float* W, const float* b,
                        const float* res, float* C, int M, int K, int Nc,
                        int do_gelu, hipStream_t s)
{
  dim3 grid((M + 63) / 64, Nc / 32);
  gemm_f32_wmma<<<grid, 256, 0, s>>>(A, W, b, res, C, M, K, Nc, do_gelu);
}

static inline long cdivl(long a, long b) { return (a + b - 1) / b; }

extern "C" void kernel_launch(void* const* d_in, const int* in_sizes, int n_in,
                              void* d_out, int out_size, void* d_ws, size_t ws_size,
                              hipStream_t stream)
{
  (void)in_sizes; (void)n_in; (void)out_size; (void)ws_size;

  // inputs in setup_inputs() order, params dict flattened in insertion order
  int idx = 0;
  const float* triplet_h = (const float*)d_in[idx++];   // 0
  const float* path_len_emb = (const float*)d_in[idx++];
  const float* virtual_emb  = (const float*)d_in[idx++];
  const float* selfloop_emb = (const float*)d_in[idx++];
  const float* dist_W1 = (const float*)d_in[idx++];
  const float* dist_b1 = (const float*)d_in[idx++];
  const float* dist_W2 = (const float*)d_in[idx++];
  const float* dist_b2 = (const float*)d_in[idx++];
  const float* trip_Wi = (const float*)d_in[idx++];
  const float* trip_bi = (const float*)d_in[idx++];
  const float* trip_Wo = (const float*)d_in[idx++];
  const float* trip_bo = (const float*)d_in[idx++];
  const float* path_W1 = (const float*)d_in[idx++];
  const float* path_b1 = (const float*)d_in[idx++];
  const float* path_W2 = (const float*)d_in[idx++];
  const float* path_b2 = (const float*)d_in[idx++];
  const float* attn_ln_g = (const float*)d_in[idx++];
  const float* attn_ln_b = (const float*)d_in[idx++];
  const float* qkv_W = (const float*)d_in[idx++];
  const float* qkv_b = (const float*)d_in[idx++];
  const float* res_ln_g = (const float*)d_in[idx++];
  const float* res_ln_b = (const float*)d_in[idx++];
  const float* res_in_W = (const float*)d_in[idx++];
  const float* res_in_b = (const float*)d_in[idx++];
  const float* ffn_W0 = (const float*)d_in[idx++];
  const float* ffn_b0 = (const float*)d_in[idx++];
  const float* ffn_W1 = (const float*)d_in[idx++];
  const float* ffn_b1 = (const float*)d_in[idx++];
  const float* ffn_W2 = (const float*)d_in[idx++];
  const float* ffn_b2 = (const float*)d_in[idx++];
  const float* ffn_W3 = (const float*)d_in[idx++];
  const float* ffn_b3 = (const float*)d_in[idx++];
  const int* src  = (const int*)d_in[idx++];
  const int* dst  = (const int*)d_in[idx++];
  const int* path = (const int*)d_in[idx++];
  const int* vp   = (const int*)d_in[idx++];
  const int* sl   = (const int*)d_in[idx++];

  // ---- workspace carving (floats) ----
  float* ws = (float*)d_ws;
  size_t off = 0;
  float* path_attn = ws + off; off += (size_t)E_EDGES * N_HEADS;  // persists
  int*   sel       = (int*)(ws + off); off += E_EDGES;            // persists
  float* dist8     = ws + off; off += 64;                         // persists
  float* S = ws + off;  // shared arena, aliased between phases

  // phase 1 layout
  float* t_all    = S;
  float* trip_tmp = t_all + (size_t)L_PATH * (N_NODES + 1) * DT_DIM;
  float* path_h   = trip_tmp + (size_t)N_NODES * DT_DIM;
  float* ph1      = path_h + (size_t)E_EDGES * DT_DIM;

  // phase 2 layout (aliases phase 1 region; phase-1 data dead by then)
  float* hbuf  = S;
  float* hn    = hbuf + (size_t)N_NODES * D_MODEL;
  float* qkv   = hn + (size_t)N_NODES * D_MODEL;
  float* logit = qkv + (size_t)N_NODES * 3 * D_MODEL;
  float* mbuf  = logit + (size_t)E_EDGES * N_HEADS;
  float* den   = mbuf + (size_t)N_NODES * N_HEADS;
  float* agg   = den + (size_t)N_NODES * N_HEADS;
  float* xbuf  = agg + (size_t)N_NODES * D_MODEL;
  float* y0    = xbuf + (size_t)N_NODES * D_MODEL;
  float* y1    = y0 + (size_t)N_NODES * DFF_DIM;

  // ================= phase 1: attention-bias precompute =================
  dist8_kernel<<<1, 128, 0, stream>>>(path_len_emb, virtual_emb, selfloop_emb,
                                      dist_W1, dist_b1, dist_W2, dist_b2, dist8);
  sel_kernel<<<cdivl(E_EDGES, 256), 256, 0, stream>>>(path, vp, sl, sel);

  long t_all_n = (long)L_PATH * (N_NODES + 1) * DT_DIM;
  fill_f32<<<cdivl(t_all_n, 256), 256, 0, stream>>>(t_all, 0.0f, t_all_n);

  for (int l = 0; l < L_PATH; ++l) {
    gemm(triplet_h, trip_Wi + (size_t)l * D_MODEL * DT_DIM, trip_bi + l * DT_DIM,
         nullptr, trip_tmp, N_NODES, D_MODEL, DT_DIM, 1, stream);
    gemm(trip_tmp, trip_Wo + (size_t)l * DT_DIM * DT_DIM, trip_bo + l * DT_DIM,
         nullptr, t_all + (size_t)l * (N_NODES + 1) * DT_DIM,
         N_NODES, DT_DIM, DT_DIM, 0, stream);
  }
  path_gather_kernel<<<cdivl(E_EDGES, 8), 256, 0, stream>>>(path, t_all, path_h);
  gemm(path_h, path_W1, path_b1, nullptr, ph1, E_EDGES, DT_DIM, DT_DIM, 1, stream);
  path_attn_kernel<<<cdivl((long)E_EDGES * N_HEADS, 256), 256, 0, stream>>>(
      ph1, path_W2, path_b2, path_attn);

  // ================= phase 2: transformer layers =================
  copy_f32<<<cdivl((long)N_NODES * D_MODEL, 256), 256, 0, stream>>>(
      triplet_h, hbuf, (long)N_NODES * D_MODEL);

  for (int i = 0; i < NLAYERS; ++i) {
    const float* qW = qkv_W + (size_t)i * D_MODEL * 3 * D_MODEL;
    const float* qB = qkv_b + (size_t)i * 3 * D_MODEL;

    ln_kernel<<<cdivl(N_NODES, 8), 256, 0, stream>>>(
        hbuf, attn_ln_g + i * D_MODEL, attn_ln_b + i * D_MODEL, hn, N_NODES);
    gemm(hn, qW, qB, nullptr, qkv, N_NODES, D_MODEL, 3 * D_MODEL, 0, stream);

    long nh = (long)N_NODES * N_HEADS;
    fill_f32<<<cdivl(nh, 256), 256, 0, stream>>>(mbuf, -INFINITY, nh);
    fill_f32<<<cdivl(nh, 256), 256, 0, stream>>>(den, 0.0f, nh);
    fill_f32<<<cdivl((long)N_NODES * D_MODEL, 256), 256, 0, stream>>>(
        agg, 0.0f, (long)N_NODES * D_MODEL);

    long eh = (long)E_EDGES * N_HEADS;
    logits_kernel<<<cdivl(eh, 256), 256, 0, stream>>>(
        qkv, src, dst, sel, dist8, path_attn, logit);
    segmax_kernel<<<cdivl(eh, 256), 256, 0, stream>>>(logit, dst, mbuf);
    exps_kernel<<<cdivl(eh, 256), 256, 0, stream>>>(logit, dst, mbuf, den);
    agg_kernel<<<cdivl((long)E_EDGES * D_MODEL, 256), 256, 0, stream>>>(
        logit, den, qkv, src, dst, agg);

    // x = h + agg @ res_in_W + b
    gemm(agg, res_in_W + (size_t)i * D_MODEL * D_MODEL, res_in_b + i * D_MODEL,
         hbuf, xbuf, N_NODES, D_MODEL, D_MODEL, 0, stream);

    ln_kernel<<<cdivl(N_NODES, 8), 256, 0, stream>>>(
        xbuf, res_ln_g + i * D_MODEL, res_ln_b + i * D_MODEL, hn, N_NODES);

    gemm(hn, ffn_W0 + (size_t)i * D_MODEL * DFF_DIM, ffn_b0 + i * DFF_DIM,
         nullptr, y0, N_NODES, D_MODEL, DFF_DIM, 1, stream);
    gemm(y0, ffn_W1 + (size_t)i * DFF_DIM * DFF_DIM, ffn_b1 + i * DFF_DIM,
         nullptr, y1, N_NODES, DFF_DIM, DFF_DIM, 1, stream);
    gemm(y1, ffn_W2 + (size_t)i * DFF_DIM * DFF_DIM, ffn_b2 + i * DFF_DIM,
         nullptr, y0, N_NODES, DFF_DIM, DFF_DIM, 1, stream);

    float* hout = (i == NLAYERS - 1) ? (float*)d_out : hbuf;
    gemm(y0, ffn_W3 + (size_t)i * DFF_DIM * D_MODEL, ffn_b3 + i * D_MODEL,
         xbuf, hout, N_NODES, DFF_DIM, D_MODEL, 0, stream);
  }
}